// S4FixSideChainModel_14293651161723
// MI455X (gfx1250) — compile-verified
//
#include <hip/hip_runtime.h>

// S4FixSideChain on MI455X (gfx1250, wave32).
// B=32, L=32768, H=32, N=32 complex modes, 3 DSSM layers, 2 Linear(32,32).
// DSSM = 3-phase chunked parallel scan (chunk S=512, NC=64).
// Linears = V_WMMA_F32_16X16X4_F32; LDS staging via GLOBAL_LOAD_ASYNC_TO_LDS_B128.

#define B_ 32
#define L_ 32768
#define H_ 32
#define N_ 32
#define S_ 512
#define NC_ 64
#define NLAYER 3
#define WPAD 40   // LDS row stride (floats): 160B, 16B-aligned, half-waves 16 banks apart

typedef __attribute__((ext_vector_type(2))) float v2f;
typedef __attribute__((ext_vector_type(8))) float v8f;

__device__ __forceinline__ float gelu_f(float v) {
    // jax.nn.gelu default: tanh approximation
    float v3 = v * v * v;
    return 0.5f * v * (1.0f + tanhf(0.7978845608028654f * (v + 0.044715f * v3)));
}

// Async DMA: 16B global -> LDS, tracked on ASYNCcnt (no VGPR data path).
__device__ __forceinline__ void async_b128(void* lds_dst, const void* gsrc) {
    uint32_t loff = (uint32_t)(uintptr_t)lds_dst;  // low 32 bits of generic = LDS addr
    asm volatile("global_load_async_to_lds_b128 %0, %1, off"
                 :: "v"(loff), "v"(gsrc) : "memory");
}
__device__ __forceinline__ void wait_async0() {
    asm volatile("s_wait_asynccnt 0x0" ::: "memory");
}

// ---------------------------------------------------------------------------
// Precompute per-(layer,h,n): lambda = exp(dt*A), C' = 2*C*expm1(dt*A)/A,
// lambda^S for carry propagation.
// ---------------------------------------------------------------------------
__global__ void k_params(const float* __restrict__ log_dt,
                         const float* __restrict__ C_re,
                         const float* __restrict__ C_im,
                         const float* __restrict__ lAr,
                         const float* __restrict__ Aim,
                         float4* __restrict__ lamc2,
                         float2* __restrict__ lamS) {
    int i = blockIdx.x * blockDim.x + threadIdx.x;
    if (i >= NLAYER * H_ * N_) return;
    int layer = i / (H_ * N_);
    int hn = i - layer * H_ * N_;
    int h = hn / N_;
    float dt = expf(log_dt[layer * H_ + h]);
    float Are = -expf(lAr[i]);
    float Ai  = Aim[i];
    float re = dt * Are, im = dt * Ai;
    float er = expf(re);
    float lr = er * cosf(im), li = er * sinf(im);
    // expm1(dtA)/A = ((lr-1) + i*li) * conj(A) / |A|^2
    float den = Are * Are + Ai * Ai;
    float fr = ((lr - 1.f) * Are + li * Ai) / den;
    float fi = (li * Are - (lr - 1.f) * Ai) / den;
    float cr = C_re[i], ci = C_im[i];
    float c2r = 2.f * (cr * fr - ci * fi);
    float c2i = 2.f * (cr * fi + ci * fr);
    float esr = expf((float)S_ * re);
    float sim = (float)S_ * im;
    lamc2[i] = make_float4(lr, li, c2r, c2i);
    lamS[i]  = make_float2(esr * cosf(sim), esr * sinf(sim));
}

// ---------------------------------------------------------------------------
// Phase 1: per (b,h,chunk), lanes = modes. Async-stage chunk to LDS, scan from
// zero state, emit end state. Input affine (stem) folded into the scan step.
// ---------------------------------------------------------------------------
__global__ void __launch_bounds__(32)
k_scan_local(const float* __restrict__ uIn, const float* __restrict__ x,
             const float* __restrict__ w_in, const float* __restrict__ b_in,
             int stem, const float4* __restrict__ lamc2,
             float2* __restrict__ states) {
    int c = blockIdx.x, h = blockIdx.y, b = blockIdx.z;
    int n = threadIdx.x;
    __shared__ float uS[S_];
    const float* base = stem ? (x + (size_t)b * L_ + c * S_)
                             : (uIn + ((size_t)(b * H_ + h)) * L_ + c * S_);
#pragma unroll
    for (int i = 0; i < S_ / 128; ++i) {  // 4 x b128 per lane, coalesced
        int f4 = i * 32 + n;
        async_b128(&uS[f4 * 4], base + f4 * 4);
    }
    wait_async0();
    __syncthreads();
    float wi = stem ? w_in[h] : 1.f;
    float bi = stem ? b_in[h] : 0.f;
    float4 p = lamc2[h * N_ + n];
    float zr = 0.f, zi = 0.f;
#pragma unroll 8
    for (int t = 0; t < S_; ++t) {
        float u = fmaf(uS[t], wi, bi);  // broadcast LDS read, conflict-free
        float nzr = fmaf(p.x, zr, fmaf(-p.y, zi, u));
        zi = fmaf(p.x, zi, p.y * zr);
        zr = nzr;
    }
    states[((size_t)(c * B_ + b) * H_ + h) * N_ + n] = make_float2(zr, zi);
}

// ---------------------------------------------------------------------------
// Phase 2: per (b,h,n), sequential combine over 64 chunks with lambda^S.
// ---------------------------------------------------------------------------
__global__ void k_scan_carry(const float2* __restrict__ states,
                             const float2* __restrict__ lamS,
                             float2* __restrict__ carry) {
    int i = blockIdx.x * blockDim.x + threadIdx.x;
    if (i >= B_ * H_ * N_) return;
    int b = i / (H_ * N_);
    int hn = i - b * H_ * N_;
    float2 ls = lamS[hn];
    float cr = 0.f, ci = 0.f;
    for (int c = 0; c < NC_; ++c) {
        size_t idx = (size_t)(c * B_ + b) * (H_ * N_) + hn;
        carry[idx] = make_float2(cr, ci);
        float2 ze = states[idx];
        float ncr = fmaf(ls.x, cr, fmaf(-ls.y, ci, ze.x));
        ci = fmaf(ls.x, ci, fmaf(ls.y, cr, ze.y));
        cr = ncr;
    }
}

// ---------------------------------------------------------------------------
// Phase 3: per (b,h) block, lane = chunk. All 32 complex states in VGPRs,
// params broadcast from LDS; fused y = Re(C'z)+D*u and GELU.
// ---------------------------------------------------------------------------
__global__ void __launch_bounds__(NC_)
k_scan_final(const float* __restrict__ uIn, const float* __restrict__ x,
             const float* __restrict__ w_in, const float* __restrict__ b_in,
             int stem, const float4* __restrict__ lamc2,
             const float2* __restrict__ carry, const float* __restrict__ Dsk,
             float* __restrict__ actOut) {
    int h = blockIdx.x, b = blockIdx.y;
    int c = threadIdx.x;
    __shared__ float4 pS[N_];
    if (threadIdx.x < N_) pS[threadIdx.x] = lamc2[h * N_ + threadIdx.x];
    float zr[N_], zi[N_];
#pragma unroll
    for (int n = 0; n < N_; ++n) {
        float2 cy = carry[(size_t)(c * B_ + b) * (H_ * N_) + h * N_ + n];
        zr[n] = cy.x; zi[n] = cy.y;
    }
    __syncthreads();
    float Dh = Dsk[h];
    float wi = stem ? w_in[h] : 0.f;
    float bi = stem ? b_in[h] : 0.f;
    const float* ub = stem ? (x + (size_t)b * L_ + c * S_)
                           : (uIn + ((size_t)(b * H_ + h)) * L_ + c * S_);
    float* yb = actOut + ((size_t)(b * H_ + h)) * L_ + c * S_;
    for (int t0 = 0; t0 < S_; t0 += 4) {
        float4 u4 = *(const float4*)(ub + t0);
        if (t0 + 64 < S_) __builtin_prefetch(ub + t0 + 64, 0, 0);  // global_prefetch
        if (stem) {
            u4.x = fmaf(u4.x, wi, bi); u4.y = fmaf(u4.y, wi, bi);
            u4.z = fmaf(u4.z, wi, bi); u4.w = fmaf(u4.w, wi, bi);
        }
        float uu[4] = {u4.x, u4.y, u4.z, u4.w};
        float oo[4];
#pragma unroll
        for (int j = 0; j < 4; ++j) {
            float u = uu[j];
            float acc = Dh * u;
#pragma unroll
            for (int n = 0; n < N_; ++n) {
                float4 p = pS[n];
                float nzr = fmaf(p.x, zr[n], fmaf(-p.y, zi[n], u));
                zi[n] = fmaf(p.x, zi[n], p.y * zr[n]);
                zr[n] = nzr;
                acc = fmaf(p.z, zr[n], fmaf(-p.w, zi[n], acc));
            }
            oo[j] = gelu_f(acc);
        }
        *(float4*)(yb + t0) = make_float4(oo[0], oo[1], oo[2], oo[3]);
    }
}

// ---------------------------------------------------------------------------
// Linear(H=32 -> K=32) over (B,H,L) via V_WMMA_F32_16X16X4_F32.
// Block: one b, 32 l-positions; 4 waves cover (ktile x ltile) 16x16 outputs.
// Tiles staged via async b128 DMA; LDS row stride 40 floats (conflict-free).
// ---------------------------------------------------------------------------
__global__ void __launch_bounds__(128)
k_linear_wmma(const float* __restrict__ in, const float* __restrict__ W,
              const float* __restrict__ bias, float* __restrict__ out) {
    int b = blockIdx.y;
    int l0 = blockIdx.x * 32;
    int tid = threadIdx.x;
    __shared__ float Ws[H_ * WPAD];  // [h][k], stride 40
    __shared__ float As[H_ * WPAD];  // [h][l], stride 40
    // 32 rows x 8 chunks of 16B each = 256 chunks for each tile; 2 per thread.
#pragma unroll
    for (int i = 0; i < 2; ++i) {
        int chunk = tid + i * 128;
        int row = chunk >> 3, seg = chunk & 7;
        async_b128(&Ws[row * WPAD + seg * 4], W + row * H_ + seg * 4);
        async_b128(&As[row * WPAD + seg * 4],
                   in + ((size_t)(b * H_ + row)) * L_ + l0 + seg * 4);
    }
    wait_async0();
    __syncthreads();
    int wave = tid >> 5, lane = tid & 31;
    int kt = wave & 1, lt = wave >> 1;     // output k-tile / l-tile
    int m16 = lane & 15, hi = lane >> 4;
    v8f acc;
#pragma unroll
    for (int j = 0; j < 8; ++j) acc[j] = bias[kt * 16 + j + 8 * hi];
#pragma unroll
    for (int kk = 0; kk < 8; ++kk) {       // K=32 in steps of 4
        int kb = kk * 4 + hi * 2;          // lanes 16-31 carry K+2,K+3 (ISA 16x4 layout)
        v2f a, bb;
        a[0] = Ws[kb * WPAD + kt * 16 + m16];        // A[m,k] = W[h=k][kout=m]
        a[1] = Ws[(kb + 1) * WPAD + kt * 16 + m16];
        bb[0] = As[kb * WPAD + lt * 16 + m16];       // B[k,n] = act[h=k][l=n]
        bb[1] = As[(kb + 1) * WPAD + lt * 16 + m16];
        acc = __builtin_amdgcn_wmma_f32_16x16x4_f32(
            false, a, false, bb, (short)0, acc, false, false);
    }
#pragma unroll
    for (int j = 0; j < 8; ++j) {
        int m = kt * 16 + j + 8 * hi;  // C/D layout: vgpr j -> M=j (lanes<16) / j+8
        int n = lt * 16 + m16;
        out[((size_t)(b * H_ + m)) * L_ + l0 + n] = acc[j];
    }
}

// ---------------------------------------------------------------------------
// Final: out[b,l] = x[b,l] * (sum_h act[b,h,l]*w_out[h] + b_out)
// ---------------------------------------------------------------------------
__global__ void k_output(const float* __restrict__ act, const float* __restrict__ x,
                         const float* __restrict__ w_out, const float* __restrict__ b_out,
                         float* __restrict__ out) {
    int b = blockIdx.y;
    int l = blockIdx.x * 256 + threadIdx.x;
    float acc = b_out[0];
#pragma unroll
    for (int h = 0; h < H_; ++h)
        acc = fmaf(act[((size_t)(b * H_ + h)) * L_ + l], w_out[h], acc);
    size_t i = (size_t)b * L_ + l;
    out[i] = x[i] * acc;
}

extern "C" void kernel_launch(void* const* d_in, const int* in_sizes, int n_in,
                              void* d_out, int out_size, void* d_ws, size_t ws_size,
                              hipStream_t stream) {
    const float* x      = (const float*)d_in[0];
    const float* w_in   = (const float*)d_in[1];
    const float* b_in   = (const float*)d_in[2];
    const float* w_mid  = (const float*)d_in[3];
    const float* b_mid  = (const float*)d_in[4];
    const float* w_out  = (const float*)d_in[5];
    const float* b_out  = (const float*)d_in[6];
    const float* log_dt = (const float*)d_in[7];
    const float* C_re   = (const float*)d_in[8];
    const float* C_im   = (const float*)d_in[9];
    const float* lAr    = (const float*)d_in[10];
    const float* Aim    = (const float*)d_in[11];
    const float* Dsk    = (const float*)d_in[12];
    float* outp = (float*)d_out;

    char* ws = (char*)d_ws;
    size_t actBytes = (size_t)B_ * H_ * L_ * sizeof(float);        // 128 MB
    size_t stBytes  = (size_t)NC_ * B_ * H_ * N_ * sizeof(float2); // 16 MB
    float*  bufA   = (float*)ws;  ws += actBytes;
    float*  bufU   = (float*)ws;  ws += actBytes;
    float2* states = (float2*)ws; ws += stBytes;
    float2* carry  = (float2*)ws; ws += stBytes;
    float4* lamc2  = (float4*)ws; ws += (size_t)NLAYER * H_ * N_ * sizeof(float4);
    float2* lamS   = (float2*)ws; ws += (size_t)NLAYER * H_ * N_ * sizeof(float2);

    k_params<<<(NLAYER * H_ * N_ + 255) / 256, 256, 0, stream>>>(
        log_dt, C_re, C_im, lAr, Aim, lamc2, lamS);

    dim3 g1(NC_, H_, B_);
    dim3 g3(H_, B_);
    int nb2 = (B_ * H_ * N_ + 255) / 256;

    // Stem DSSM (input affine fused) -> bufA (post-GELU)
    k_scan_local<<<g1, 32, 0, stream>>>(nullptr, x, w_in, b_in, 1, lamc2, states);
    k_scan_carry<<<nb2, 256, 0, stream>>>(states, lamS, carry);
    k_scan_final<<<g3, NC_, 0, stream>>>(nullptr, x, w_in, b_in, 1, lamc2, carry,
                                         Dsk, bufA);
    // Depth blocks: Linear (WMMA) -> DSSM -> GELU
    for (int i = 0; i < 2; ++i) {
        k_linear_wmma<<<dim3(L_ / 32, B_), 128, 0, stream>>>(
            bufA, w_mid + (size_t)i * H_ * H_, b_mid + (size_t)i * H_, bufU);
        int lay = i + 1;
        k_scan_local<<<g1, 32, 0, stream>>>(bufU, nullptr, nullptr, nullptr, 0,
                                            lamc2 + (size_t)lay * H_ * N_, states);
        k_scan_carry<<<nb2, 256, 0, stream>>>(states, lamS + (size_t)lay * H_ * N_,
                                              carry);
        k_scan_final<<<g3, NC_, 0, stream>>>(bufU, nullptr, nullptr, nullptr, 0,
                                             lamc2 + (size_t)lay * H_ * N_, carry,
                                             Dsk + lay * H_, bufA);
    }
    // Output gain + modulation
    k_output<<<dim3(L_ / 256, B_), 256, 0, stream>>>(bufA, x, w_out, b_out, outp);
}